// VisualImitation_hard_83588653514800
// MI455X (gfx1250) — compile-verified
//
#include <hip/hip_runtime.h>

// ---------------------------------------------------------------------------
// Scatter: img[clip(floor(x*2048))*2048 + clip(floor(y*2048))] = z
// Deterministic last-index-wins, SINGLE pass over the 100.7 MB input:
//   ws[idx] = atomicMax_u64(ws[idx], ((p+1) << 32) | bits(z))
// then unpack ws -> img. Guaranteed HBM traffic ~151 MB => ~6.5 us @ 23.3 TB/s.
// ---------------------------------------------------------------------------

#define IMG_DIM   2048
#define IMG_DIMF  2048.0f

typedef float              vf4 __attribute__((ext_vector_type(4)));
typedef unsigned           vu4 __attribute__((ext_vector_type(4)));
typedef unsigned long long u64;
typedef u64                vul2 __attribute__((ext_vector_type(2)));

__device__ __forceinline__ int vi_bin(float v) {
    int i = (int)floorf(v * IMG_DIMF);
    i = i < 0 ? 0 : i;
    i = i > (IMG_DIM - 1) ? (IMG_DIM - 1) : i;
    return i;
}

__device__ __forceinline__ u64 vi_pack(unsigned rank, float z) {
    return ((u64)rank << 32) | (u64)__float_as_uint(z);
}

// ===================== u64 single-pass path =================================

// Zero the u64 winner workspace with b128 stores (2 pixels / lane).
__global__ __launch_bounds__(256)
void vi_fill64(u64* __restrict__ ws, int count) {
    int t    = blockIdx.x * blockDim.x + threadIdx.x;
    int base = t * 2;
    if (base + 1 < count) {
        vul2 z = {0ull, 0ull};
        *(vul2*)(ws + base) = z;                      // global_store_b128
    } else if (base < count) {
        ws[base] = 0ull;
    }
}

// One pass over the points: 4 points / lane via 3 coalesced NT b128 loads,
// 4 x global_atomic_max_u64 into the L2-resident workspace.
__global__ __launch_bounds__(256)
void vi_scatter64(const float* __restrict__ x, u64* __restrict__ ws, int n) {
    int t  = blockIdx.x * blockDim.x + threadIdx.x;
    int p0 = t * 4;
    if (p0 >= n) return;

    const vf4* x4 = (const vf4*)x;
    // Prefetch the input stream ~24 KB ahead (gfx1250 global_prefetch_b8).
    __builtin_prefetch(x4 + (size_t)3 * t + 1536, 0, 0);

    if (p0 + 3 < n) {
        vf4 a = __builtin_nontemporal_load(x4 + (size_t)3 * t + 0);
        vf4 b = __builtin_nontemporal_load(x4 + (size_t)3 * t + 1);
        vf4 c = __builtin_nontemporal_load(x4 + (size_t)3 * t + 2);
        unsigned i0 = (unsigned)(vi_bin(a.x) * IMG_DIM + vi_bin(a.y));
        unsigned i1 = (unsigned)(vi_bin(a.w) * IMG_DIM + vi_bin(b.x));
        unsigned i2 = (unsigned)(vi_bin(b.z) * IMG_DIM + vi_bin(b.w));
        unsigned i3 = (unsigned)(vi_bin(c.y) * IMG_DIM + vi_bin(c.z));
        atomicMax(&ws[i0], vi_pack((unsigned)p0 + 1u, a.z));  // global_atomic_max_u64
        atomicMax(&ws[i1], vi_pack((unsigned)p0 + 2u, b.y));
        atomicMax(&ws[i2], vi_pack((unsigned)p0 + 3u, c.x));
        atomicMax(&ws[i3], vi_pack((unsigned)p0 + 4u, c.w));
    } else {
        for (int p = p0; p < n; ++p) {
            unsigned idx = (unsigned)(vi_bin(x[3 * p]) * IMG_DIM + vi_bin(x[3 * p + 1]));
            atomicMax(&ws[idx], vi_pack((unsigned)p + 1u, x[3 * p + 2]));
        }
    }
}

// Unpack: every pixel written exactly once (no separate output zeroing).
// ws stream is dead after this -> NT loads.
__global__ __launch_bounds__(256)
void vi_unpack64(const u64* __restrict__ ws, float* __restrict__ out, int count) {
    int t    = blockIdx.x * blockDim.x + threadIdx.x;
    int base = t * 4;
    if (base + 3 < count) {
        vul2 w01 = __builtin_nontemporal_load((const vul2*)(ws + base));
        vul2 w23 = __builtin_nontemporal_load((const vul2*)(ws + base + 2));
        vf4 o;
        o.x = (w01.x >> 32) ? __uint_as_float((unsigned)w01.x) : 0.0f;
        o.y = (w01.y >> 32) ? __uint_as_float((unsigned)w01.y) : 0.0f;
        o.z = (w23.x >> 32) ? __uint_as_float((unsigned)w23.x) : 0.0f;
        o.w = (w23.y >> 32) ? __uint_as_float((unsigned)w23.y) : 0.0f;
        *(vf4*)(out + base) = o;                      // global_store_b128
    } else {
        for (int k = base; k < count; ++k) {
            u64 w  = ws[k];
            out[k] = (w >> 32) ? __uint_as_float((unsigned)w) : 0.0f;
        }
    }
}

// ===================== u32 two-pass fallback ================================

__global__ __launch_bounds__(256)
void vi_zero(float* __restrict__ out, unsigned* __restrict__ win,
             int out_size, int do_win) {
    int t    = blockIdx.x * blockDim.x + threadIdx.x;
    int base = t * 4;
    if (base + 3 < out_size) {
        vf4 zf = {0.0f, 0.0f, 0.0f, 0.0f};
        vu4 zu = {0u, 0u, 0u, 0u};
        *(vf4*)(out + base) = zf;
        if (do_win) *(vu4*)(win + base) = zu;
    } else {
        for (int k = base; k < out_size; ++k) {
            out[k] = 0.0f;
            if (do_win) win[k] = 0u;
        }
    }
}

__global__ __launch_bounds__(256)
void vi_pass1(const float* __restrict__ x, unsigned* __restrict__ win, int n) {
    int t  = blockIdx.x * blockDim.x + threadIdx.x;
    int p0 = t * 4;
    if (p0 >= n) return;
    const vf4* x4 = (const vf4*)x;
    __builtin_prefetch(x4 + (size_t)3 * t + 1536, 0, 0);
    if (p0 + 3 < n) {
        vf4 a = x4[(size_t)3 * t + 0];
        vf4 b = x4[(size_t)3 * t + 1];
        vf4 c = x4[(size_t)3 * t + 2];
        unsigned i0 = (unsigned)(vi_bin(a.x) * IMG_DIM + vi_bin(a.y));
        unsigned i1 = (unsigned)(vi_bin(a.w) * IMG_DIM + vi_bin(b.x));
        unsigned i2 = (unsigned)(vi_bin(b.z) * IMG_DIM + vi_bin(b.w));
        unsigned i3 = (unsigned)(vi_bin(c.y) * IMG_DIM + vi_bin(c.z));
        atomicMax(&win[i0], (unsigned)p0 + 1u);
        atomicMax(&win[i1], (unsigned)p0 + 2u);
        atomicMax(&win[i2], (unsigned)p0 + 3u);
        atomicMax(&win[i3], (unsigned)p0 + 4u);
    } else {
        for (int p = p0; p < n; ++p) {
            unsigned idx = (unsigned)(vi_bin(x[3 * p]) * IMG_DIM + vi_bin(x[3 * p + 1]));
            atomicMax(&win[idx], (unsigned)p + 1u);
        }
    }
}

__global__ __launch_bounds__(256)
void vi_pass2(const float* __restrict__ x, const unsigned* __restrict__ win,
              float* __restrict__ out, int n) {
    int t  = blockIdx.x * blockDim.x + threadIdx.x;
    int p0 = t * 4;
    if (p0 >= n) return;
    const vf4* x4 = (const vf4*)x;
    if (p0 + 3 < n) {
        vf4 a = __builtin_nontemporal_load(x4 + (size_t)3 * t + 0);
        vf4 b = __builtin_nontemporal_load(x4 + (size_t)3 * t + 1);
        vf4 c = __builtin_nontemporal_load(x4 + (size_t)3 * t + 2);
        unsigned i0 = (unsigned)(vi_bin(a.x) * IMG_DIM + vi_bin(a.y));
        unsigned i1 = (unsigned)(vi_bin(a.w) * IMG_DIM + vi_bin(b.x));
        unsigned i2 = (unsigned)(vi_bin(b.z) * IMG_DIM + vi_bin(b.w));
        unsigned i3 = (unsigned)(vi_bin(c.y) * IMG_DIM + vi_bin(c.z));
        if (win[i0] == (unsigned)p0 + 1u) out[i0] = a.z;
        if (win[i1] == (unsigned)p0 + 2u) out[i1] = b.y;
        if (win[i2] == (unsigned)p0 + 3u) out[i2] = c.x;
        if (win[i3] == (unsigned)p0 + 4u) out[i3] = c.w;
    } else {
        for (int p = p0; p < n; ++p) {
            unsigned idx = (unsigned)(vi_bin(x[3 * p]) * IMG_DIM + vi_bin(x[3 * p + 1]));
            if (win[idx] == (unsigned)p + 1u) out[idx] = x[3 * p + 2];
        }
    }
}

// ===================== last-resort plain scatter ============================

__global__ __launch_bounds__(256)
void vi_scatter_plain(const float* __restrict__ x, float* __restrict__ out, int n) {
    int t  = blockIdx.x * blockDim.x + threadIdx.x;
    int p0 = t * 4;
    if (p0 >= n) return;
    const vf4* x4 = (const vf4*)x;
    if (p0 + 3 < n) {
        vf4 a = x4[(size_t)3 * t + 0];
        vf4 b = x4[(size_t)3 * t + 1];
        vf4 c = x4[(size_t)3 * t + 2];
        out[vi_bin(a.x) * IMG_DIM + vi_bin(a.y)] = a.z;
        out[vi_bin(a.w) * IMG_DIM + vi_bin(b.x)] = b.y;
        out[vi_bin(b.z) * IMG_DIM + vi_bin(b.w)] = c.x;
        out[vi_bin(c.y) * IMG_DIM + vi_bin(c.z)] = c.w;
    } else {
        for (int p = p0; p < n; ++p)
            out[vi_bin(x[3 * p]) * IMG_DIM + vi_bin(x[3 * p + 1])] = x[3 * p + 2];
    }
}

// ===================== host launcher ========================================

extern "C" void kernel_launch(void* const* d_in, const int* in_sizes, int n_in,
                              void* d_out, int out_size, void* d_ws, size_t ws_size,
                              hipStream_t stream) {
    const float* x   = (const float*)d_in[0];
    float*       out = (float*)d_out;

    const int n      = in_sizes[0] / 3;              // number of points
    const int nq     = (n + 3) / 4;                  // 4 points per lane
    const int blocks = (nq + 255) / 256;

    if (ws_size >= (size_t)out_size * sizeof(u64)) {
        // --- single-pass u64 winner path (preferred) ---
        u64* ws = (u64*)d_ws;
        int fthreads = (out_size + 1) / 2;           // 2 pixels / lane
        vi_fill64<<<(fthreads + 255) / 256, 256, 0, stream>>>(ws, out_size);
        vi_scatter64<<<blocks, 256, 0, stream>>>(x, ws, n);
        int uthreads = (out_size + 3) / 4;           // 4 pixels / lane
        vi_unpack64<<<(uthreads + 255) / 256, 256, 0, stream>>>(ws, out, out_size);
    } else if (ws_size >= (size_t)out_size * sizeof(unsigned)) {
        // --- two-pass u32 winner path ---
        unsigned* win = (unsigned*)d_ws;
        int zthreads  = (out_size + 3) / 4;
        vi_zero<<<(zthreads + 255) / 256, 256, 0, stream>>>(out, win, out_size, 1);
        vi_pass1<<<blocks, 256, 0, stream>>>(x, win, n);
        vi_pass2<<<blocks, 256, 0, stream>>>(x, win, out, n);
    } else {
        // --- no workspace: plain scatter (in-bounds, winner unspecified) ---
        int zthreads = (out_size + 3) / 4;
        vi_zero<<<(zthreads + 255) / 256, 256, 0, stream>>>(out, (unsigned*)d_ws, out_size, 0);
        vi_scatter_plain<<<blocks, 256, 0, stream>>>(x, out, n);
    }
}